// VRFC_5669356831750
// MI455X (gfx1250) — compile-verified
//
#include <hip/hip_runtime.h>
#include <hip/hip_bf16.h>

typedef __attribute__((ext_vector_type(2))) float v2f;
typedef __attribute__((ext_vector_type(8))) float v8f;

#define N_OBJ      512
#define N_OBJ_CLS  151
#define N_REL_CLS  51
#define REL_DIM    4096
#define N_REL      65536
#define NMS_TH     0.3f

// ---------------------------------------------------------------------------
// Kernel 1: row softmax (512 x 151) + copy logits to output. 1 wave per row.
// ---------------------------------------------------------------------------
__global__ void softmax_copy_kernel(const float* __restrict__ logits,
                                    float* __restrict__ out_dists,
                                    float* __restrict__ probs) {
    const int lane = threadIdx.x & 31;
    const int wv   = threadIdx.x >> 5;
    const int row  = blockIdx.x * 8 + wv;          // grid 64 * 8 waves = 512 rows
    const float* rp = logits + (size_t)row * N_OBJ_CLS;

    float v[5];
    float mx = -3.4e38f;
#pragma unroll
    for (int j = 0; j < 5; ++j) {
        int c = lane + 32 * j;
        v[j] = (c < N_OBJ_CLS) ? rp[c] : -3.4e38f;
        if (c < N_OBJ_CLS) {
            out_dists[(size_t)row * N_OBJ_CLS + c] = v[j];   // obj_dists2 passthrough
            mx = fmaxf(mx, v[j]);
        }
    }
#pragma unroll
    for (int off = 16; off > 0; off >>= 1) mx = fmaxf(mx, __shfl_xor(mx, off, 32));

    float sum = 0.0f;
#pragma unroll
    for (int j = 0; j < 5; ++j) {
        int c = lane + 32 * j;
        if (c < N_OBJ_CLS) { v[j] = __expf(v[j] - mx); sum += v[j]; }
    }
#pragma unroll
    for (int off = 16; off > 0; off >>= 1) sum += __shfl_xor(sum, off, 32);

    const float inv = 1.0f / sum;
#pragma unroll
    for (int j = 0; j < 5; ++j) {
        int c = lane + 32 * j;
        if (c < N_OBJ_CLS) probs[(size_t)row * N_OBJ_CLS + c] = v[j] * inv;
    }
}

// ---------------------------------------------------------------------------
// Kernel 2: per-class NMS. One block (512 threads) per class c in [0,150).
// Bitonic argsort (descending) of scores in LDS, then greedy sweep.
// ---------------------------------------------------------------------------
__global__ void nms_kernel(const float* __restrict__ probs,
                           const float* __restrict__ boxes,  // (512,151,4)
                           float* __restrict__ mask) {       // (512,150)
    const int c = blockIdx.x;       // class index into [1..150] is c+1
    const int t = threadIdx.x;      // 0..511

    __shared__ float skey[N_OBJ];
    __shared__ int   sidx[N_OBJ];
    __shared__ float bx1[N_OBJ], by1[N_OBJ], bx2[N_OBJ], by2[N_OBJ];
    __shared__ int   active[N_OBJ];

    skey[t] = probs[(size_t)t * N_OBJ_CLS + (c + 1)];
    sidx[t] = t;
    __syncthreads();

    // Bitonic sort, descending by score
    for (int k = 2; k <= N_OBJ; k <<= 1) {
        for (int j = k >> 1; j > 0; j >>= 1) {
            int ixj = t ^ j;
            if (ixj > t) {
                bool dirDesc = ((t & k) == 0);
                float a = skey[t], b = skey[ixj];
                bool sw = dirDesc ? (a < b) : (a > b);
                if (sw) {
                    skey[t] = b; skey[ixj] = a;
                    int tmp = sidx[t]; sidx[t] = sidx[ixj]; sidx[ixj] = tmp;
                }
            }
            __syncthreads();
        }
    }

    const int oi = sidx[t];
    const float* bp = boxes + ((size_t)oi * N_OBJ_CLS + (c + 1)) * 4;
    const float x1 = bp[0], y1 = bp[1], x2 = bp[2], y2 = bp[3];
    bx1[t] = x1; by1[t] = y1; bx2[t] = x2; by2[t] = y2;
    active[t] = 1;
    const float areaT = (x2 - x1 + 1.0f) * (y2 - y1 + 1.0f);
    __syncthreads();

    // Greedy sweep: pivot i suppresses only indices > i, so reading active[i]
    // races with nothing in iteration i; active[t] is frozen after iter t.
    for (int i = 0; i < N_OBJ; ++i) {
        if (active[i] && t > i) {
            float ix1 = bx1[i], iy1 = by1[i], ix2 = bx2[i], iy2 = by2[i];
            float xx1 = fmaxf(ix1, x1), yy1 = fmaxf(iy1, y1);
            float xx2 = fminf(ix2, x2), yy2 = fminf(iy2, y2);
            float w = fmaxf(xx2 - xx1 + 1.0f, 0.0f);
            float h = fmaxf(yy2 - yy1 + 1.0f, 0.0f);
            float inter = w * h;
            float areaI = (ix2 - ix1 + 1.0f) * (iy2 - iy1 + 1.0f);
            float iou = inter / (areaT + areaI - inter);
            if (iou > NMS_TH) active[t] = 0;
        }
        __syncthreads();
    }

    mask[(size_t)oi * 150 + c] = active[t] ? 1.0f : 0.0f;
}

// ---------------------------------------------------------------------------
// Kernel 3: obj_preds = argmax_c (mask * probs[:,1:]) + 1  (first-max wins)
// ---------------------------------------------------------------------------
__global__ void argmax_kernel(const float* __restrict__ probs,
                              const float* __restrict__ mask,
                              float* __restrict__ preds) {
    const int o = blockIdx.x * 256 + threadIdx.x;
    if (o >= N_OBJ) return;
    float best = -1.0f; int bi = 0;
    for (int cc = 0; cc < 150; ++cc) {
        float v = mask[(size_t)o * 150 + cc] * probs[(size_t)o * N_OBJ_CLS + cc + 1];
        if (v > best) { best = v; bi = cc; }
    }
    preds[o] = (float)(bi + 1);
}

// ---------------------------------------------------------------------------
// Kernel 4: zero-pad w (51x4096) to wpad (64x4096) so B-loads are branch-free
// ---------------------------------------------------------------------------
__global__ void padw_kernel(const float* __restrict__ w, float* __restrict__ wpad) {
    const int idx = blockIdx.x * 256 + threadIdx.x;   // 0 .. 64*4096-1
    const int row = idx >> 12;
    wpad[idx] = (row < N_REL_CLS) ? w[idx] : 0.0f;
}

// ---------------------------------------------------------------------------
// Kernel 5: rel_dists = vr @ w.T + b via V_WMMA_F32_16X16X4_F32.
// Block = 128 threads (4 waves). Each wave: 32 rows x 64 padded cols.
// A-frag f32 16x4 layout: lanes 0-15 hold K={0,1}, lanes 16-31 hold K={2,3}
// (contiguous pair per lane -> one b64 load). B-frag mirrors it with rows = N.
// ---------------------------------------------------------------------------
__global__ void __launch_bounds__(128)
gemm_wmma_kernel(const float* __restrict__ vr,
                 const float* __restrict__ wpad,   // 64 x 4096, zero padded
                 const float* __restrict__ bias,   // 51
                 float* __restrict__ out) {        // 65536 x 51
    const int lane = threadIdx.x & 31;
    const int wv   = threadIdx.x >> 5;
    const int l15  = lane & 15;
    const int hi   = lane >> 4;                 // 0 or 1
    const int m0   = blockIdx.x * 128 + wv * 32;
    const int kh   = hi * 2;                    // K sub-offset per A/B layout

    const float* a0 = vr   + (size_t)(m0 + l15) * REL_DIM + kh;
    const float* a1 = a0   + (size_t)16 * REL_DIM;
    const float* b0 = wpad + (size_t)l15 * REL_DIM + kh;
    const float* b1 = b0   + (size_t)16 * REL_DIM;
    const float* b2 = b0   + (size_t)32 * REL_DIM;
    const float* b3 = b0   + (size_t)48 * REL_DIM;

    v8f acc[2][4] = {};

    for (int kk = 0; kk < REL_DIM; kk += 32) {
        // stream-ahead prefetch of the two vr rows this lane owns
        __builtin_prefetch(a0 + kk + 512, 0, 1);
        __builtin_prefetch(a1 + kk + 512, 0, 1);
#pragma unroll
        for (int ku = 0; ku < 32; ku += 4) {
            const int k = kk + ku;
            v2f A0 = *(const v2f*)(a0 + k);
            v2f A1 = *(const v2f*)(a1 + k);
            v2f B0 = *(const v2f*)(b0 + k);
            v2f B1 = *(const v2f*)(b1 + k);
            v2f B2 = *(const v2f*)(b2 + k);
            v2f B3 = *(const v2f*)(b3 + k);
            acc[0][0] = __builtin_amdgcn_wmma_f32_16x16x4_f32(false, A0, false, B0, (short)0, acc[0][0], false, false);
            acc[0][1] = __builtin_amdgcn_wmma_f32_16x16x4_f32(false, A0, false, B1, (short)0, acc[0][1], false, false);
            acc[0][2] = __builtin_amdgcn_wmma_f32_16x16x4_f32(false, A0, false, B2, (short)0, acc[0][2], false, false);
            acc[0][3] = __builtin_amdgcn_wmma_f32_16x16x4_f32(false, A0, false, B3, (short)0, acc[0][3], false, false);
            acc[1][0] = __builtin_amdgcn_wmma_f32_16x16x4_f32(false, A1, false, B0, (short)0, acc[1][0], false, false);
            acc[1][1] = __builtin_amdgcn_wmma_f32_16x16x4_f32(false, A1, false, B1, (short)0, acc[1][1], false, false);
            acc[1][2] = __builtin_amdgcn_wmma_f32_16x16x4_f32(false, A1, false, B2, (short)0, acc[1][2], false, false);
            acc[1][3] = __builtin_amdgcn_wmma_f32_16x16x4_f32(false, A1, false, B3, (short)0, acc[1][3], false, false);
        }
    }

    // Epilogue: C/D layout — VGPR j holds row (j + 8*hi), col = lane&15
#pragma unroll
    for (int t = 0; t < 2; ++t) {
#pragma unroll
        for (int n = 0; n < 4; ++n) {
            const int col = n * 16 + l15;
            if (col < N_REL_CLS) {
                const float bv = bias[col];
#pragma unroll
                for (int j = 0; j < 8; ++j) {
                    const int row = m0 + t * 16 + j + 8 * hi;
                    out[(size_t)row * N_REL_CLS + col] = acc[t][n][j] + bv;
                }
            }
        }
    }
}

// ---------------------------------------------------------------------------
extern "C" void kernel_launch(void* const* d_in, const int* in_sizes, int n_in,
                              void* d_out, int out_size, void* d_ws, size_t ws_size,
                              hipStream_t stream) {
    const float* obj_logits = (const float*)d_in[0];   // 512*151
    const float* vr         = (const float*)d_in[1];   // 65536*4096
    const float* boxes      = (const float*)d_in[2];   // 512*151*4
    const float* w          = (const float*)d_in[3];   // 51*4096
    const float* b          = (const float*)d_in[4];   // 51

    float* out       = (float*)d_out;
    float* out_dists = out;                         // 512*151 = 77312
    float* out_preds = out + 77312;                 // 512
    float* out_rel   = out + 77824;                 // 65536*51

    float* ws    = (float*)d_ws;
    float* probs = ws;                              // 512*151 = 77312 floats
    float* mask  = ws + 77312;                      // 512*150 = 76800 floats
    float* wpad  = ws + 154112;                     // 64*4096 = 262144 floats

    softmax_copy_kernel<<<64, 256, 0, stream>>>(obj_logits, out_dists, probs);
    nms_kernel<<<150, 512, 0, stream>>>(probs, boxes, mask);
    argmax_kernel<<<2, 256, 0, stream>>>(probs, mask, out_preds);
    padw_kernel<<<1024, 256, 0, stream>>>(w, wpad);
    gemm_wmma_kernel<<<512, 128, 0, stream>>>(vr, wpad, b, out_rel);
}